// AirsSpectralGNN_6416681140925
// MI455X (gfx1250) — compile-verified
//
#include <hip/hip_runtime.h>
#include <hip/hip_bf16.h>
#include <math.h>

// ---------------------------------------------------------------------------
// Problem constants
// ---------------------------------------------------------------------------
#define BV      16
#define CV      16384
#define NROWS   (BV * CV)          // 262144 rows of (b,c)
#define HIDV    128
#define SA      132                // padded LDS row stride (dwords) for A tiles
#define HBYTES  ((size_t)NROWS * HIDV * sizeof(float))   // 128 MB

typedef float v2f __attribute__((ext_vector_type(2)));
typedef float v8f __attribute__((ext_vector_type(8)));

__device__ __forceinline__ float gelu_exact(float v) {
  return 0.5f * v * (1.0f + erff(v * 0.70710678118654752f));
}

__device__ __forceinline__ float wave_sum32(float v) {
#pragma unroll
  for (int off = 16; off > 0; off >>= 1) v += __shfl_xor(v, off, 32);
  return v;
}

// ---------------------------------------------------------------------------
// Wave-level 16x16 output tile, K=128, fp32 WMMA (V_WMMA_F32_16X16X4_F32).
// A tile: LDS row-major [16][sa].  B: LDS row-major 128x128.  n0: column base.
// A layout (16x4, 32-bit): lanes 0-15 = M, VGPR0/1 = K0/K1 (lanes 0-15),
//                          K2/K3 (lanes 16-31).
// B layout (4x16, 32-bit): lanes = N, VGPR0/1 = K0/K1 (lanes 0-15),
//                          K2/K3 (lanes 16-31).
// C/D: VGPR i -> M = i (+8 for lanes 16-31), N = lane&15.
// ---------------------------------------------------------------------------
__device__ __forceinline__ v8f wmma_k128_f32(const float* __restrict__ As, int sa,
                                             const float* __restrict__ Bs, int n0,
                                             int lane) {
  const int half = (lane >> 4) & 1;
  const int l16  = lane & 15;
  v8f acc = {};
#pragma unroll
  for (int k0 = 0; k0 < HIDV; k0 += 4) {
    const int ks = k0 + half * 2;
    v2f a = *(const v2f*)(As + l16 * sa + ks);   // 8B aligned: sa even, ks even
    v2f b;
    b.x = Bs[(ks + 0) * HIDV + n0 + l16];
    b.y = Bs[(ks + 1) * HIDV + n0 + l16];
    acc = __builtin_amdgcn_wmma_f32_16x16x4_f32(false, a, false, b,
                                                (short)0, acc, false, false);
  }
  return acc;
}

// ---------------------------------------------------------------------------
// Kernel 1: build graph.  Single workgroup bitonic argsort of wavelengths in
// LDS (64KB keys + 64KB indices), then per-node stencil neighbors + weights.
// deg = 2 at chain endpoints, 3 inside; dinv = deg^-1/2.
// ---------------------------------------------------------------------------
__global__ void __launch_bounds__(1024)
graph_build_kernel(const float* __restrict__ wl,
                   int* __restrict__ nbrp, int* __restrict__ nbrn,
                   float* __restrict__ wp, float* __restrict__ wsf,
                   float* __restrict__ wn) {
  extern __shared__ char smem[];
  float* keys = (float*)smem;                 // [16384]
  int*   idx  = (int*)(smem + CV * sizeof(float));
  const int tid = threadIdx.x;

  for (int i = tid; i < CV; i += 1024) { keys[i] = wl[i]; idx[i] = i; }
  __syncthreads();

  for (int k = 2; k <= CV; k <<= 1) {
    for (int j = k >> 1; j > 0; j >>= 1) {
#pragma unroll 1
      for (int p = 0; p < CV / 1024; ++p) {
        int i  = tid + p * 1024;
        int ij = i ^ j;
        if (ij > i) {
          bool  up = ((i & k) == 0);
          float ki = keys[i], kj = keys[ij];
          bool  sw = up ? (ki > kj) : (ki < kj);
          if (sw) {
            keys[i] = kj; keys[ij] = ki;
            int t = idx[i]; idx[i] = idx[ij]; idx[ij] = t;
          }
        }
      }
      __syncthreads();
    }
  }

  // dinv per node (reuse keys[] storage, indexed by node id)
  for (int r = tid; r < CV; r += 1024) {
    int n = idx[r];
    float deg = (r == 0 || r == CV - 1) ? 2.0f : 3.0f;
    keys[n] = rsqrtf(deg);
  }
  __syncthreads();

  for (int r = tid; r < CV; r += 1024) {
    int n  = idx[r];
    int np = (r > 0) ? idx[r - 1] : -1;
    int nn = (r < CV - 1) ? idx[r + 1] : -1;
    float di = keys[n];
    nbrp[n] = np;
    nbrn[n] = nn;
    wp[n]  = (np >= 0) ? di * keys[np] : 0.0f;
    wn[n]  = (nn >= 0) ? di * keys[nn] : 0.0f;
    wsf[n] = di * di;
  }
}

// ---------------------------------------------------------------------------
// Kernel 2: fused input MLP: h = gelu(x @ in_w1 + b1) @ in_w2 + b2.
// 256 threads / 8 waves, 128 rows per block (8 tiles of 16 rows).
// First GEMM (K=8) on VALU into the LDS A-tile; second via WMMA.
// ---------------------------------------------------------------------------
__global__ void __launch_bounds__(256)
input_mlp_kernel(const float* __restrict__ x,
                 const float* __restrict__ w1, const float* __restrict__ b1,
                 const float* __restrict__ w2, const float* __restrict__ b2,
                 float* __restrict__ out) {
  extern __shared__ char smem[];
  float* W1s = (float*)smem;                           // 8*128 floats
  float* W2s = (float*)(smem + 4096);                  // 128*128 floats
  float* As  = (float*)(smem + 4096 + 65536);          // 16*SA floats

  const int tid  = threadIdx.x;
  const int lane = tid & 31;
  const int wave = tid >> 5;
  const int n0   = wave * 16;
  const int l16  = lane & 15;
  const int half = lane >> 4;

  for (int i = tid; i < 8 * HIDV; i += 256) W1s[i] = w1[i];
  {
    const float4* src = (const float4*)w2;
    float4*       dst = (float4*)W2s;
    for (int i = tid; i < (HIDV * HIDV) / 4; i += 256) dst[i] = src[i];
  }
  __syncthreads();

  const long rowBase = (long)blockIdx.x * 128;
#pragma unroll 1
  for (int it = 0; it < 8; ++it) {
    const long row0 = rowBase + it * 16;

    // build gelu(x@w1+b1) rows in LDS: thread -> (row tid>>4, 8 cols)
    {
      const int  r  = tid >> 4;
      const int  jb = (tid & 15) * 8;
      const long gr = row0 + r;
      const float4* xr4 = (const float4*)(x + gr * 8);
      float4 x0 = xr4[0], x1 = xr4[1];
      float xr[8] = {x0.x, x0.y, x0.z, x0.w, x1.x, x1.y, x1.z, x1.w};
#pragma unroll
      for (int jj = 0; jj < 8; ++jj) {
        const int j = jb + jj;
        float s = b1[j];
#pragma unroll
        for (int f = 0; f < 8; ++f) s = fmaf(xr[f], W1s[f * HIDV + j], s);
        As[r * SA + j] = gelu_exact(s);
      }
    }
    __syncthreads();

    v8f acc = wmma_k128_f32(As, SA, W2s, n0, lane);
    const float bn = b2[n0 + l16];
#pragma unroll
    for (int i = 0; i < 8; ++i) {
      out[(row0 + i + half * 8) * HIDV + n0 + l16] = acc[i] + bn;
    }
    __syncthreads();
  }
}

// ---------------------------------------------------------------------------
// Kernel 3: generic 128x128 GEMM  out = A @ W + bias  (per-layer GCN matmul).
// ---------------------------------------------------------------------------
__global__ void __launch_bounds__(256)
gemm128_kernel(const float* __restrict__ A, const float* __restrict__ W,
               const float* __restrict__ bias, float* __restrict__ out) {
  extern __shared__ char smem[];
  float* Ws = (float*)smem;                  // 128*128 floats
  float* As = (float*)(smem + 65536);        // 16*SA floats

  const int tid  = threadIdx.x;
  const int lane = tid & 31;
  const int wave = tid >> 5;
  const int n0   = wave * 16;
  const int l16  = lane & 15;
  const int half = lane >> 4;

  {
    const float4* src = (const float4*)W;
    float4*       dst = (float4*)Ws;
    for (int i = tid; i < (HIDV * HIDV) / 4; i += 256) dst[i] = src[i];
  }
  __syncthreads();

  const long rowBase = (long)blockIdx.x * 128;
#pragma unroll 1
  for (int it = 0; it < 8; ++it) {
    const long row0 = rowBase + it * 16;
    {
      const int r  = tid >> 4;
      const int cb = (tid & 15) * 8;
      const float4* src = (const float4*)(A + (row0 + r) * HIDV + cb);
      float4* dst = (float4*)(As + r * SA + cb);
      dst[0] = src[0];
      dst[1] = src[1];
    }
    __syncthreads();

    v8f acc = wmma_k128_f32(As, SA, Ws, n0, lane);
    const float bn = bias[n0 + l16];
#pragma unroll
    for (int i = 0; i < 8; ++i) {
      out[(row0 + i + half * 8) * HIDV + n0 + l16] = acc[i] + bn;
    }
    __syncthreads();
  }
}

// ---------------------------------------------------------------------------
// Kernel 4: GCN message stencil + residual + LayerNorm + ReLU (in-place on h).
// One wave per (b,c) row; 4 floats per lane (wave32 x 4 = 128).
// ---------------------------------------------------------------------------
__global__ void __launch_bounds__(256)
gcn_stencil_kernel(const float* __restrict__ hl, float* __restrict__ h,
                   const int* __restrict__ nbrp, const int* __restrict__ nbrn,
                   const float* __restrict__ wp, const float* __restrict__ wsf,
                   const float* __restrict__ wn,
                   const float* __restrict__ g, const float* __restrict__ beta) {
  const long grow = (long)blockIdx.x * 8 + (threadIdx.x >> 5);
  const int  lane = threadIdx.x & 31;
  const int  b    = (int)(grow >> 14);        // / CV
  const int  c    = (int)(grow & (CV - 1));

  const float4 hs = ((const float4*)(hl + grow * HIDV))[lane];
  const float4 hr = ((const float4*)(h  + grow * HIDV))[lane];
  const float  Ws = wsf[c], Wp = wp[c], Wn = wn[c];
  const int    np = nbrp[c], nn = nbrn[c];

  float4 a;
  a.x = hr.x + Ws * hs.x;  a.y = hr.y + Ws * hs.y;
  a.z = hr.z + Ws * hs.z;  a.w = hr.w + Ws * hs.w;
  if (np >= 0) {
    const float4 p = ((const float4*)(hl + ((long)b * CV + np) * HIDV))[lane];
    a.x += Wp * p.x; a.y += Wp * p.y; a.z += Wp * p.z; a.w += Wp * p.w;
  }
  if (nn >= 0) {
    const float4 q = ((const float4*)(hl + ((long)b * CV + nn) * HIDV))[lane];
    a.x += Wn * q.x; a.y += Wn * q.y; a.z += Wn * q.z; a.w += Wn * q.w;
  }

  const float mu = wave_sum32(a.x + a.y + a.z + a.w) * (1.0f / 128.0f);
  float4 d;
  d.x = a.x - mu; d.y = a.y - mu; d.z = a.z - mu; d.w = a.w - mu;
  const float var =
      wave_sum32(d.x * d.x + d.y * d.y + d.z * d.z + d.w * d.w) * (1.0f / 128.0f);
  const float inv = rsqrtf(var + 1e-5f);

  const float4 gg = ((const float4*)g)[lane];
  const float4 bb = ((const float4*)beta)[lane];
  float4 o;
  o.x = fmaxf(0.0f, d.x * inv * gg.x + bb.x);
  o.y = fmaxf(0.0f, d.y * inv * gg.y + bb.y);
  o.z = fmaxf(0.0f, d.z * inv * gg.z + bb.z);
  o.w = fmaxf(0.0f, d.w * inv * gg.w + bb.w);
  ((float4*)(h + grow * HIDV))[lane] = o;
}

// ---------------------------------------------------------------------------
// Kernel 5: head.  LN -> WMMA GEMM (head_w1) + gelu -> 128->2 proj + clip.
// ---------------------------------------------------------------------------
__global__ void __launch_bounds__(256)
head_kernel(const float* __restrict__ h,
            const float* __restrict__ lng, const float* __restrict__ lnb,
            const float* __restrict__ w1, const float* __restrict__ b1,
            const float* __restrict__ w2, const float* __restrict__ b2,
            float* __restrict__ out) {
  extern __shared__ char smem[];
  float* W1s = (float*)smem;                          // 128*128 floats
  float* As  = (float*)(smem + 65536);                // 16*SA floats
  float* Zs  = (float*)(smem + 65536 + 16 * SA * 4);  // 16*SA floats

  const int tid  = threadIdx.x;
  const int lane = tid & 31;
  const int wave = tid >> 5;
  const int n0   = wave * 16;
  const int l16  = lane & 15;
  const int half = lane >> 4;

  {
    const float4* src = (const float4*)w1;
    float4*       dst = (float4*)W1s;
    for (int i = tid; i < (HIDV * HIDV) / 4; i += 256) dst[i] = src[i];
  }
  __syncthreads();

  const long rowBase = (long)blockIdx.x * 128;
#pragma unroll 1
  for (int it = 0; it < 8; ++it) {
    const long row0 = rowBase + it * 16;

    // LayerNorm 16 rows into As: 8 waves x 2 rows
#pragma unroll
    for (int rr = 0; rr < 2; ++rr) {
      const int  r  = wave * 2 + rr;
      const long gr = row0 + r;
      const float4 v = ((const float4*)(h + gr * HIDV))[lane];
      const float mu = wave_sum32(v.x + v.y + v.z + v.w) * (1.0f / 128.0f);
      float4 d;
      d.x = v.x - mu; d.y = v.y - mu; d.z = v.z - mu; d.w = v.w - mu;
      const float var =
          wave_sum32(d.x * d.x + d.y * d.y + d.z * d.z + d.w * d.w) * (1.0f / 128.0f);
      const float inv = rsqrtf(var + 1e-5f);
      const float4 gg = ((const float4*)lng)[lane];
      const float4 bb = ((const float4*)lnb)[lane];
      float4 o;
      o.x = d.x * inv * gg.x + bb.x;
      o.y = d.y * inv * gg.y + bb.y;
      o.z = d.z * inv * gg.z + bb.z;
      o.w = d.w * inv * gg.w + bb.w;
      ((float4*)(As + r * SA))[lane] = o;
    }
    __syncthreads();

    v8f acc = wmma_k128_f32(As, SA, W1s, n0, lane);
    const float bn = b1[n0 + l16];
#pragma unroll
    for (int i = 0; i < 8; ++i) {
      Zs[(i + half * 8) * SA + n0 + l16] = gelu_exact(acc[i] + bn);
    }
    __syncthreads();

    if (tid < 32) {
      const int r  = tid >> 1;
      const int oc = tid & 1;
      float s = b2[oc];
#pragma unroll 4
      for (int k = 0; k < HIDV; ++k) s = fmaf(Zs[r * SA + k], w2[k * 2 + oc], s);
      if (oc == 1) s = fminf(3.0f, fmaxf(-7.0f, s));
      out[(row0 + r) * 2 + oc] = s;
    }
    __syncthreads();
  }
}

// ---------------------------------------------------------------------------
// Host launcher
// ---------------------------------------------------------------------------
extern "C" void kernel_launch(void* const* d_in, const int* in_sizes, int n_in,
                              void* d_out, int out_size, void* d_ws, size_t ws_size,
                              hipStream_t stream) {
  (void)in_sizes; (void)n_in; (void)out_size; (void)ws_size;

  const float* x        = (const float*)d_in[0];
  const float* wl       = (const float*)d_in[1];
  const float* in_w1    = (const float*)d_in[2];
  const float* in_b1    = (const float*)d_in[3];
  const float* in_w2    = (const float*)d_in[4];
  const float* in_b2    = (const float*)d_in[5];
  const float* gcn_w    = (const float*)d_in[6];
  const float* gcn_b    = (const float*)d_in[7];
  const float* gcn_g    = (const float*)d_in[8];
  const float* gcn_beta = (const float*)d_in[9];
  const float* h_ln_g   = (const float*)d_in[10];
  const float* h_ln_b   = (const float*)d_in[11];
  const float* h_w1     = (const float*)d_in[12];
  const float* h_b1     = (const float*)d_in[13];
  const float* h_w2     = (const float*)d_in[14];
  const float* h_b2     = (const float*)d_in[15];

  char*  ws    = (char*)d_ws;
  float* hbuf  = (float*)ws;                       // N x 128
  float* hlbuf = (float*)(ws + HBYTES);            // N x 128
  char*  gbase = ws + 2 * HBYTES;
  int*   nbrp  = (int*)(gbase);
  int*   nbrn  = (int*)(gbase + 1 * CV * 4);
  float* wp    = (float*)(gbase + 2 * CV * 4);
  float* wsf   = (float*)(gbase + 3 * CV * 4);
  float* wn    = (float*)(gbase + 4 * CV * 4);

  // 1) graph build: single-WG bitonic argsort in 128KB LDS
  graph_build_kernel<<<1, 1024, 2 * CV * 4, stream>>>(wl, nbrp, nbrn, wp, wsf, wn);

  // 2) fused input MLP -> h
  const int gemmBlocks = NROWS / 128;  // 2048
  input_mlp_kernel<<<gemmBlocks, 256, 4096 + 65536 + 16 * SA * 4, stream>>>(
      x, in_w1, in_b1, in_w2, in_b2, hbuf);

  // 3) 4 GCN layers: WMMA GEMM then stencil+LN+ReLU (in place on h)
  for (int l = 0; l < 4; ++l) {
    gemm128_kernel<<<gemmBlocks, 256, 65536 + 16 * SA * 4, stream>>>(
        hbuf, gcn_w + (size_t)l * HIDV * HIDV, gcn_b + (size_t)l * HIDV, hlbuf);
    gcn_stencil_kernel<<<NROWS / 8, 256, 0, stream>>>(
        hlbuf, hbuf, nbrp, nbrn, wp, wsf, wn,
        gcn_g + (size_t)l * HIDV, gcn_beta + (size_t)l * HIDV);
  }

  // 4) head: LN -> GEMM+gelu -> 128->2 projection + clip -> out (B,C,2)
  head_kernel<<<gemmBlocks, 256, 65536 + 2 * 16 * SA * 4, stream>>>(
      hbuf, h_ln_g, h_ln_b, h_w1, h_b1, h_w2, h_b2, (float*)d_out);
}